// SimpleGATSingleHeadLayerEdgeReprFeat_83476984365540
// MI455X (gfx1250) — compile-verified
//
#include <hip/hip_runtime.h>
#include <cstddef>

// ---------------------------------------------------------------------------
// SimpleGATSingleHeadLayer for MI455X (gfx1250, wave32, WMMA f32 16x16x4)
//
// d_in order: h[N,64], e[E,64], Wh[64,64], We[64,64], Wproj[192,64],
//             bproj[64], Wattn[192], src[E], dst[E]
// d_out: h_out[N,64] ++ e_proj[E,64]
// ws:    z_h[N*64] ++ ex[E] ++ denom[N]   (all f32)
//
// DRAM roofline ~31us @ 23.3 TB/s => bandwidth-bound: fp32 WMMA, one pass
// over the 328MB edge arrays, LDS-resident pre-paired weights (1 ds_b64 per
// B-fragment, conflict-free), register-resident A-fragments, persistent
// blocks. An empty `asm volatile` opaque-zero inside the tile loop makes the
// LDS weight base loop-variant so LICM cannot hoist the 192 invariant
// ds_load_b64s into registers (round-2 spills) -- without volatile loads
// (which round-3 showed lower to SCOPE_SYS flat_loads).
// ---------------------------------------------------------------------------

typedef __attribute__((ext_vector_type(2))) float v2f;
typedef __attribute__((ext_vector_type(8))) float v8f;

#define FD 64

static __device__ __forceinline__ v8f wmma4(v2f a, v2f b, v8f c) {
  // V_WMMA_F32_16X16X4_F32 : D = A(16x4) * B(4x16) + C(16x16)
  return __builtin_amdgcn_wmma_f32_16x16x4_f32(
      false, a, false, b, (short)0, c, false, false);
}

// ---- LDS layout constants (in floats) -------------------------------------
#define PS 80                 // pair-row stride in float2 units (==16 mod 32)
#define WE_F   0                      // We pairs:    32*PS*2 = 5120 floats
#define WP_F   (32 * PS * 2)          // Wproj pairs: 96*PS*2 = 15360 floats
#define WA_F   (WP_F + 96 * PS * 2)   // Wattn: 192
#define WB_F   (WA_F + 192)           // bproj: 64
#define ZE_F   (WB_F + 64)            // z_e slabs: WPB * 16*68
#define ZE_STRIDE 68                  // padded row stride (floats)
#define ZE_SLAB   (16 * ZE_STRIDE)    // 1088 floats per wave
#define WPB 4                         // waves per block (edge kernel)
#define EDGE_LDS_FLOATS (ZE_F + WPB * ZE_SLAB)

// Zero h_out (N*64) and denom (N).
__global__ void gat_init_kernel(float* __restrict__ hout,
                                float* __restrict__ denom, int N) {
  int i = blockIdx.x * blockDim.x + threadIdx.x;
  if (i < N * FD) hout[i] = 0.0f;
  if (i < N)      denom[i] = 0.0f;
}

// ---------------------------------------------------------------------------
// z_h = h @ Wh : grid-stride over 16-row tiles, Wh pre-paired in static LDS.
// ---------------------------------------------------------------------------
__global__ void gat_zh_kernel(const float* __restrict__ h,
                              const float* __restrict__ Wh,
                              float* __restrict__ zh, int nTiles) {
  __shared__ __align__(16) float WhP[32 * PS * 2];   // 20 KB
  const int t = threadIdx.x;
  for (int i = t; i < 32 * 64; i += 256) {
    int p = i >> 6, c = i & 63;
    WhP[(p * PS + c) * 2    ] = Wh[(2 * p    ) * FD + c];
    WhP[(p * PS + c) * 2 + 1] = Wh[(2 * p + 1) * FD + c];
  }
  __syncthreads();

  const int lane = t & 31, half = lane >> 4, idx = lane & 15;
  // wave-uniform values in SGPRs -> scalar loop control, EXEC untouched
  const int wslot  = __builtin_amdgcn_readfirstlane(t >> 5);
  const int wpb    = blockDim.x >> 5;
  const int stride = gridDim.x * wpb;

  for (int tile = blockIdx.x * wpb + wslot; tile < nTiles; tile += stride) {
    // Opaque zero: makes the LDS weight base loop-variant (blocks LICM).
    int opq = 0;
    asm volatile("" : "+v"(opq));
    const float* wlds = WhP + opq;

    const int row0 = tile * 16;
    const float* arow = h + (size_t)(row0 + idx) * FD;
    v2f av[16];
#pragma unroll
    for (int kt = 0; kt < 16; ++kt)
      av[kt] = *(const v2f*)(arow + kt * 4 + half * 2);

    v8f acc[4] = {};
#pragma unroll
    for (int kt = 0; kt < 16; ++kt) {
      const float* bp = wlds + ((kt * 2 + half) * PS) * 2;
#pragma unroll
      for (int nt = 0; nt < 4; ++nt)
        acc[nt] = wmma4(av[kt], *(const v2f*)(bp + (nt * 16 + idx) * 2),
                        acc[nt]);
    }
#pragma unroll
    for (int nt = 0; nt < 4; ++nt)
#pragma unroll
      for (int g = 0; g < 8; ++g)
        zh[(size_t)(row0 + g + half * 8) * FD + nt * 16 + idx] = acc[nt][g];
  }
}

// ---------------------------------------------------------------------------
// Fused per-edge kernel: z_e=e@We ; e_proj=[z_e|zs|zd]@Wproj+b ; s=z@Wattn ;
// ex=exp(s) ; denom[dst]+=ex.  16 edges per wave, persistent grid-stride.
// ---------------------------------------------------------------------------
__global__ void gat_edge_kernel(const float* __restrict__ e,
                                const float* __restrict__ We,
                                const float* __restrict__ Wproj,
                                const float* __restrict__ bproj,
                                const float* __restrict__ Wattn,
                                const int*   __restrict__ src,
                                const int*   __restrict__ dst,
                                const float* __restrict__ zh,
                                float* __restrict__ eproj,
                                float* __restrict__ exw,
                                float* __restrict__ denom,
                                int nTiles) {
  extern __shared__ __align__(16) float lds[];
  const int t = threadIdx.x;

  // ---- stage pre-paired weights (once per block) ----
  for (int i = t; i < 32 * 64; i += WPB * 32) {          // We pairs
    int p = i >> 6, c = i & 63;
    lds[WE_F + (p * PS + c) * 2    ] = We[(2 * p    ) * FD + c];
    lds[WE_F + (p * PS + c) * 2 + 1] = We[(2 * p + 1) * FD + c];
  }
  for (int i = t; i < 96 * 64; i += WPB * 32) {          // Wproj pairs
    int p = i >> 6, c = i & 63;
    lds[WP_F + (p * PS + c) * 2    ] = Wproj[(2 * p    ) * FD + c];
    lds[WP_F + (p * PS + c) * 2 + 1] = Wproj[(2 * p + 1) * FD + c];
  }
  for (int i = t; i < 192; i += WPB * 32) lds[WA_F + i] = Wattn[i];
  for (int i = t; i < 64;  i += WPB * 32) lds[WB_F + i] = bproj[i];
  __syncthreads();

  const int lane = t & 31, half = lane >> 4, idx = lane & 15;
  const int wslot = __builtin_amdgcn_readfirstlane(t >> 5);  // SGPR, uniform
  float* zeW = lds + ZE_F + wslot * ZE_SLAB;
  const int stride = gridDim.x * WPB;

  for (int tile = blockIdx.x * WPB + wslot; tile < nTiles; tile += stride) {
    // Opaque zero: weight-base becomes loop-variant -> ds loads stay in-loop.
    int opq = 0;
    asm volatile("" : "+v"(opq));
    const float* wlds = lds + opq;

    const int eid  = tile * 16 + idx;        // edge for this lane's M-row
    const int srow = src[eid];
    const int drow = dst[eid];
    const float* srcRow = zh + (size_t)srow * FD;
    const float* dstRow = zh + (size_t)drow * FD;

    // ---- phase 1: z_e = e_tile @ We, stage to padded LDS slab ----
    {
      const float* arow = e + (size_t)eid * FD;
      v2f av[16];
#pragma unroll
      for (int kt = 0; kt < 16; ++kt)
        av[kt] = *(const v2f*)(arow + kt * 4 + half * 2);

      v8f acc[4] = {};
#pragma unroll
      for (int kt = 0; kt < 16; ++kt) {
        const float* bp = wlds + WE_F + ((kt * 2 + half) * PS) * 2;
#pragma unroll
        for (int nt = 0; nt < 4; ++nt)
          acc[nt] = wmma4(av[kt], *(const v2f*)(bp + (nt * 16 + idx) * 2),
                          acc[nt]);
      }
#pragma unroll
      for (int nt = 0; nt < 4; ++nt)
#pragma unroll
        for (int g = 0; g < 8; ++g)
          zeW[(g + half * 8) * ZE_STRIDE + nt * 16 + idx] = acc[nt][g];
    }
    // per-wave LDS slab, DS ops in-order within a wave: no barrier needed.

    // ---- hoisted A-fragments (reused across n-tiles and for the logit) ----
    v2f za[16], zs[16], zd[16];
#pragma unroll
    for (int kt = 0; kt < 16; ++kt) {
      int k = kt * 4 + half * 2;
      za[kt] = *(const v2f*)(zeW + idx * ZE_STRIDE + k);
      zs[kt] = *(const v2f*)(srcRow + k);
      zd[kt] = *(const v2f*)(dstRow + k);
    }

    // ---- phase 2: e_proj = z_e@Wp0 + zs@Wp1 + zd@Wp2 + b (K = 192) ----
    {
      v8f acc[4] = {};
#pragma unroll
      for (int kt = 0; kt < 16; ++kt) {
        const float* bp = wlds + WP_F + ((kt * 2 + half) * PS) * 2;
#pragma unroll
        for (int nt = 0; nt < 4; ++nt)
          acc[nt] = wmma4(za[kt], *(const v2f*)(bp + (nt * 16 + idx) * 2),
                          acc[nt]);
      }
#pragma unroll
      for (int kt = 0; kt < 16; ++kt) {
        const float* bp = wlds + WP_F + ((32 + kt * 2 + half) * PS) * 2;
#pragma unroll
        for (int nt = 0; nt < 4; ++nt)
          acc[nt] = wmma4(zs[kt], *(const v2f*)(bp + (nt * 16 + idx) * 2),
                          acc[nt]);
      }
#pragma unroll
      for (int kt = 0; kt < 16; ++kt) {
        const float* bp = wlds + WP_F + ((64 + kt * 2 + half) * PS) * 2;
#pragma unroll
        for (int nt = 0; nt < 4; ++nt)
          acc[nt] = wmma4(zd[kt], *(const v2f*)(bp + (nt * 16 + idx) * 2),
                          acc[nt]);
      }
#pragma unroll
      for (int nt = 0; nt < 4; ++nt) {
        float bias = wlds[WB_F + nt * 16 + idx];
#pragma unroll
        for (int g = 0; g < 8; ++g)
          eproj[(size_t)(tile * 16 + g + half * 8) * FD + nt * 16 + idx] =
              acc[nt][g] + bias;
      }
    }

    // ---- phase 3: logit from register fragments + cross-half reduce ----
    // lane(half,idx) covers k == {2*half, 2*half+1} (mod 4) of edge idx.
    {
      float part = 0.0f;
#pragma unroll
      for (int kt = 0; kt < 16; ++kt) {
        int k = kt * 4 + half * 2;
        v2f wa = *(const v2f*)(wlds + WA_F + k);
        v2f ws = *(const v2f*)(wlds + WA_F + 64 + k);
        v2f wd = *(const v2f*)(wlds + WA_F + 128 + k);
        part += za[kt].x * wa.x + za[kt].y * wa.y;
        part += zs[kt].x * ws.x + zs[kt].y * ws.y;
        part += zd[kt].x * wd.x + zd[kt].y * wd.y;
      }
      float s = part + __shfl_xor(part, 16, 32);   // sum the two k-halves
      if (half == 0) {                             // one lane per edge
        float ex = __expf(s);
        exw[eid] = ex;
        atomicAdd(denom + drow, ex);               // global_atomic_add_f32
      }
    }
  }
}

// h_out[dst] += (ex/denom[dst]) * z_h[src]; one thread per (edge, col).
__global__ void gat_scatter_kernel(const int* __restrict__ src,
                                   const int* __restrict__ dst,
                                   const float* __restrict__ zh,
                                   const float* __restrict__ exw,
                                   const float* __restrict__ denom,
                                   float* __restrict__ hout,
                                   long long total) {
  long long i = (long long)blockIdx.x * blockDim.x + threadIdx.x;
  if (i >= total) return;
  int eidx = (int)(i >> 6);
  int col  = (int)(i & 63);
  int d    = dst[eidx];
  float alpha = exw[eidx] / denom[d];
  atomicAdd(hout + (size_t)d * FD + col,
            alpha * zh[(size_t)src[eidx] * FD + col]);
}

extern "C" void kernel_launch(void* const* d_in, const int* in_sizes, int n_in,
                              void* d_out, int out_size, void* d_ws, size_t ws_size,
                              hipStream_t stream) {
  const float* h     = (const float*)d_in[0];
  const float* e     = (const float*)d_in[1];
  const float* Wh    = (const float*)d_in[2];
  const float* We    = (const float*)d_in[3];
  const float* Wproj = (const float*)d_in[4];
  const float* bproj = (const float*)d_in[5];
  const float* Wattn = (const float*)d_in[6];
  const int*   src   = (const int*)d_in[7];
  const int*   dst   = (const int*)d_in[8];

  const int N = in_sizes[0] / FD;   // 80000
  const int E = in_sizes[1] / FD;   // 1280000

  float* hout  = (float*)d_out;                    // [N,64]
  float* eproj = (float*)d_out + (size_t)N * FD;   // [E,64]

  float* zh    = (float*)d_ws;                     // N*64
  float* exw   = zh + (size_t)N * FD;              // E
  float* denom = exw + (size_t)E;                  // N

  // 1) zero h_out and denom
  {
    int total = N * FD;
    gat_init_kernel<<<(total + 255) / 256, 256, 0, stream>>>(hout, denom, N);
  }
  // 2) z_h = h @ Wh (8 waves/block, grid-stride tiles, Wh pairs in LDS)
  {
    int nTiles = N / 16;
    int blocks = (nTiles + 7) / 8;
    if (blocks > 1024) blocks = 1024;
    gat_zh_kernel<<<blocks, 256, 0, stream>>>(h, Wh, zh, nTiles);
  }
  // 3) fused per-edge WMMA kernel (persistent; ~98 KB dynamic LDS; CDNA5
  //    allows 320 KB per workgroup)
  {
    int nTiles = E / 16;
    int blocks = 1024;
    if (blocks * WPB > nTiles) blocks = (nTiles + WPB - 1) / WPB;
    size_t shmem = (size_t)EDGE_LDS_FLOATS * sizeof(float);
    gat_edge_kernel<<<blocks, WPB * 32, shmem, stream>>>(
        e, We, Wproj, bproj, Wattn, src, dst, zh, eproj, exw, denom, nTiles);
  }
  // 4) weighted scatter-sum into h_out
  {
    long long total = (long long)E * FD;
    int blocks = (int)((total + 255) / 256);
    gat_scatter_kernel<<<blocks, 256, 0, stream>>>(src, dst, zh, exw, denom,
                                                   hout, total);
  }
}